// RandomHybridRosenbrock_37709812858889
// MI455X (gfx1250) — compile-verified
//
#include <hip/hip_runtime.h>

// -------- problem constants (reference: N1 = 4097, N2 = 8192) --------------
#define ROWLEN         4096            // N1 - 1 columns per row of y / b
#define NROWS          8191            // N2 - 1 rows participate
#define NVALID         4094            // N1 - 3 valid columns per row
#define CHUNKS_PER_ROW 512             // ROWLEN / 8 eight-wide chunks
#define TOTAL_CHUNKS   (NROWS * CHUNKS_PER_ROW)   // 4,193,792
#define NBLK           2048
#define TPB            256             // 8 wave32 waves per block

typedef __attribute__((ext_vector_type(2))) float v2f;
typedef __attribute__((ext_vector_type(4))) float v4f;
typedef __attribute__((ext_vector_type(8))) float v8f;

// Pass 1: stream x and b once with b128 loads, accumulate per-thread partials,
// reduce each wave with V_WMMA_F32_16X16X4_F32 (exact f32), then LDS across waves.
__global__ __launch_bounds__(TPB) void rosen_partial(
    const float* __restrict__ x, const float* __restrict__ b,
    float* __restrict__ ws)
{
    float acc = 0.0f;

    const int stride = gridDim.x * blockDim.x;
    for (int g = (int)(blockIdx.x * blockDim.x + threadIdx.x);
         g < TOTAL_CHUNKS; g += stride)
    {
        const int r   = g >> 9;                   // chunk / CHUNKS_PER_ROW
        const int c0  = (g & 511) << 3;           // first column of this chunk
        const int idx = (r << 12) + c0;           // r*ROWLEN + c0 (fits 32-bit)

        // b used exactly once -> non-temporal streaming loads (16B aligned).
        const v4f b0 = __builtin_nontemporal_load((const v4f*)(b + idx));
        const v4f b1 = __builtin_nontemporal_load((const v4f*)(b + idx + 4));

        // Aligned b128 at x+idx == {y[c0-1], y[c0], y[c0+1], y[c0+2]} of row r,
        // because y = x+1: the shift hands us the y_prev element for free.
        const v4f xa = *(const v4f*)(x + idx);
        const v4f xb = *(const v4f*)(x + idx + 4);   // y[c0+3 .. c0+6]
        const float y7 = x[idx + 8];                 // y[c0+7]

        // roll(y,1,axis=1): column 0 wraps to y[r, ROWLEN-1] = x[idx + ROWLEN].
        const float yp = (c0 == 0) ? x[idx + ROWLEN] : xa.x;

        const float d0 = __builtin_fmaf(-yp,   yp,   xa.y);
        const float d1 = __builtin_fmaf(-xa.y, xa.y, xa.z);
        const float d2 = __builtin_fmaf(-xa.z, xa.z, xa.w);
        const float d3 = __builtin_fmaf(-xa.w, xa.w, xb.x);
        const float d4 = __builtin_fmaf(-xb.x, xb.x, xb.y);
        const float d5 = __builtin_fmaf(-xb.y, xb.y, xb.z);
        const float d6 = __builtin_fmaf(-xb.z, xb.z, xb.w);
        const float d7 = __builtin_fmaf(-xb.w, xb.w, y7);

        acc = __builtin_fmaf(b0.x * d0, d0, acc);
        acc = __builtin_fmaf(b0.y * d1, d1, acc);
        acc = __builtin_fmaf(b0.z * d2, d2, acc);
        acc = __builtin_fmaf(b0.w * d3, d3, acc);
        acc = __builtin_fmaf(b1.x * d4, d4, acc);
        acc = __builtin_fmaf(b1.y * d5, d5, acc);
        if (c0 + 6 < NVALID) {                    // only last chunk of each row
            acc = __builtin_fmaf(b1.z * d6, d6, acc);
            acc = __builtin_fmaf(b1.w * d7, d7, acc);
        }
    }

    // ---- wave32 reduction via V_WMMA_F32_16X16X4_F32 ----
    // A(16x4): lane L<16 -> A[L,0]=acc, A[L,1]=0; lane L>=16 -> A[L-16,2]=acc.
    // B(4x16) = ones  =>  D[m,n] = acc_m + acc_{m+16} for every n.
    // Lane n (n<16) holds D[0..7, n]; lane n+16 holds D[8..15, n].
    v2f af; af.x = acc;  af.y = 0.0f;
    v2f bf; bf.x = 1.0f; bf.y = 1.0f;
    v8f cf = {};
    cf = __builtin_amdgcn_wmma_f32_16x16x4_f32(
            /*neg_a=*/false, af, /*neg_b=*/false, bf,
            /*c_mod=*/(short)0, cf, /*reuse_a=*/false, /*reuse_b=*/false);

    float s = ((cf[0] + cf[1]) + (cf[2] + cf[3])) +
              ((cf[4] + cf[5]) + (cf[6] + cf[7]));
    s += __shfl_xor(s, 16, 32);        // pair (n, n+16) -> full 32-lane sum

    __shared__ float wpart[TPB / 32];
    const int lane = threadIdx.x & 31;
    const int wid  = threadIdx.x >> 5;
    if (lane == 0) wpart[wid] = s;
    __syncthreads();
    if (threadIdx.x == 0) {
        float bs = 0.0f;
#pragma unroll
        for (int i = 0; i < TPB / 32; ++i) bs += wpart[i];
        ws[blockIdx.x] = bs;           // one partial per block, fixed order
    }
}

// Pass 2: deterministic tree-reduce the NBLK partials, add scalar term.
__global__ __launch_bounds__(256) void rosen_finalize(
    const float* __restrict__ ws, int nblk,
    const float* __restrict__ x, const float* __restrict__ a,
    const float* __restrict__ mu, float* __restrict__ out)
{
    __shared__ float red[256];
    float s = 0.0f;
    for (int i = threadIdx.x; i < nblk; i += 256) s += ws[i];
    red[threadIdx.x] = s;
    __syncthreads();
#pragma unroll
    for (int off = 128; off > 0; off >>= 1) {
        if ((int)threadIdx.x < off) red[threadIdx.x] += red[threadIdx.x + off];
        __syncthreads();
    }
    if (threadIdx.x == 0) {
        const float d = x[0] - mu[0];
        out[0] = -(a[0] * d * d) - red[0];
    }
}

extern "C" void kernel_launch(void* const* d_in, const int* in_sizes, int n_in,
                              void* d_out, int out_size, void* d_ws, size_t ws_size,
                              hipStream_t stream) {
    // setup_inputs() order: x, a, b, mu — all float32.
    const float* x  = (const float*)d_in[0];
    const float* a  = (const float*)d_in[1];
    const float* b  = (const float*)d_in[2];
    const float* mu = (const float*)d_in[3];
    float* ws = (float*)d_ws;          // needs NBLK * 4 = 8 KB scratch

    rosen_partial<<<NBLK, TPB, 0, stream>>>(x, b, ws);
    rosen_finalize<<<1, 256, 0, stream>>>(ws, NBLK, x, a, mu, (float*)d_out);
}